// TRIPLETFORMER_68753836474773
// MI455X (gfx1250) — compile-verified
//
#include <hip/hip_runtime.h>
#include <cstdint>
#include <cstddef>

// Problem constants (from the reference): B=64, T=512, D=41
#define B_  64
#define T_  512
#define D_  41
#define L_  (T_ * D_)      // 20992 elements per batch row
#define OC_ (D_ + 3)       // 44 output channels per packed element

typedef float v4f __attribute__((ext_vector_type(4)));

// One workgroup per batch row; 512 threads (16 wave32 waves).
// Thread j owns timestep t=j (41 contiguous source elements).
__global__ __launch_bounds__(512) void
TRIPLETFORMER_pack_kernel(const float* __restrict__ target_x,
                          const float* __restrict__ vals,
                          const int*   __restrict__ mask,
                          float* __restrict__ out)
{
    extern __shared__ int s_mask[];      // L_ ints = 84KB dynamic LDS (row's mask)
    __shared__ int s_wsum[16];           // per-wave scan totals (16 waves)

    const int b    = blockIdx.x;
    const int tid  = threadIdx.x;
    const int lane = tid & 31;           // wave32
    const int wave = tid >> 5;

    // ---------------- Phase A: async DMA of this row's mask into LDS ----------
    // mask row is 20992 dwords = 83968 bytes. 10 rounds of b128 (512 lanes x 16B)
    // cover dwords [0, 20480); one b32 round covers the final 512 dwords.
    const char*    gbase    = (const char*)(mask + (size_t)b * L_);
    const unsigned lds_base = (unsigned)(uintptr_t)(&s_mask[0]);

    #pragma unroll
    for (int r = 0; r < 10; ++r) {
        unsigned byte_off = (unsigned)(r * 512 + tid) * 16u;
        unsigned lds_addr = lds_base + byte_off;
        asm volatile("global_load_async_to_lds_b128 %0, %1, %2"
                     :: "v"(lds_addr), "v"(byte_off), "s"(gbase)
                     : "memory");
    }
    {
        unsigned byte_off = (unsigned)(20480 + tid) * 4u;
        unsigned lds_addr = lds_base + byte_off;
        asm volatile("global_load_async_to_lds_b32 %0, %1, %2"
                     :: "v"(lds_addr), "v"(byte_off), "s"(gbase)
                     : "memory");
    }

    // While the DMA is in flight, prefetch this thread's vals row (164B).
    const float* vrow = vals + ((size_t)b * T_ + tid) * D_;
    __builtin_prefetch(vrow, 0, 0);

    // Drain this wave's async ops, then barrier so every wave's LDS data is visible.
    asm volatile("s_wait_asynccnt 0" ::: "memory");
    __syncthreads();

    // ---------------- Phase B: per-thread mask bits + count (from LDS) --------
    const int base = tid * D_;          // source index of this thread's first element
    uint64_t bits = 0;
    int cnt = 0;
    #pragma unroll
    for (int k = 0; k < D_; ++k) {
        int m = s_mask[base + k] & 1;
        cnt += m;
        bits |= (uint64_t)m << k;
    }

    // ---------------- Phase C: block-wide inclusive scan (wave32 shuffles) ----
    // 1) inclusive scan within each wave via shfl_up (registers only)
    int v = cnt;
    #pragma unroll
    for (int off = 1; off < 32; off <<= 1) {
        int n = __shfl_up(v, off, 32);
        if (lane >= off) v += n;
    }
    // 2) wave totals -> LDS, scanned by wave 0
    if (lane == 31) s_wsum[wave] = v;
    __syncthreads();
    if (tid < 16) {
        int w = s_wsum[tid];
        #pragma unroll
        for (int off = 1; off < 16; off <<= 1) {
            int n = __shfl_up(w, off, 32);
            if (tid >= off) w += n;
        }
        s_wsum[tid] = w;               // inclusive scan of wave totals
    }
    __syncthreads();
    const int wave_excl = (wave == 0) ? 0 : s_wsum[wave - 1];
    const int incl = v + wave_excl;    // block-wide inclusive prefix
    const int excl = incl - cnt;       // #observed before this thread's segment
    const int N1   = s_wsum[15];       // total observed in this row

    // ---------------- Phase D: scatter 44-channel output rows -----------------
    // Stable partition destinations:
    //   observed  source i -> excl_ones(i)
    //   padded    source i -> N1 + (i - excl_ones(i))
    const float tau = target_x[(size_t)b * T_ + tid];
    int p_obs = excl;
    int p_pad = N1 + base - excl;

    float* const orow_base = out + (size_t)b * L_ * OC_;

    #pragma unroll 1
    for (int k = 0; k < D_; ++k) {
        const int   m  = (int)((bits >> k) & 1);
        const float U  = vrow[k];
        const int   p  = m ? p_obs : p_pad;
        p_obs += m;
        p_pad += 1 - m;

        // Row layout (44 ch): [0]=tau_p, [1..41]=one_hot(C_p), [42]=U_p, [43]=valid
        // Padded rows: C_p==0 -> one_hot puts 1.0 at channel 1 (matches reference).
        const float v0  = m ? tau : 0.0f;
        const float v42 = m ? U   : 0.0f;
        const float v43 = m ? 1.0f : 0.0f;
        const int   oh  = m ? (k + 1) : 1;   // channel holding the 1.0 (in [1,41])

        float* const orow = orow_base + (size_t)p * OC_;  // 176B, 16B-aligned

        #pragma unroll
        for (int c4 = 0; c4 < 11; ++c4) {
            const int c = c4 * 4;
            v4f ch;
            ch.x = (c + 0 == 0)  ? v0
                 : (c + 0 == 42) ? v42
                 : (c + 0 == 43) ? v43
                 : ((c + 0 == oh) ? 1.0f : 0.0f);
            ch.y = (c + 1 == 0)  ? v0
                 : (c + 1 == 42) ? v42
                 : (c + 1 == 43) ? v43
                 : ((c + 1 == oh) ? 1.0f : 0.0f);
            ch.z = (c + 2 == 0)  ? v0
                 : (c + 2 == 42) ? v42
                 : (c + 2 == 43) ? v43
                 : ((c + 2 == oh) ? 1.0f : 0.0f);
            ch.w = (c + 3 == 0)  ? v0
                 : (c + 3 == 42) ? v42
                 : (c + 3 == 43) ? v43
                 : ((c + 3 == oh) ? 1.0f : 0.0f);
            // Output (236MB) is write-once, never re-read: non-temporal b128 store.
            __builtin_nontemporal_store(ch, (v4f*)(orow + c));
        }
    }
}

extern "C" void kernel_launch(void* const* d_in, const int* in_sizes, int n_in,
                              void* d_out, int out_size, void* d_ws, size_t ws_size,
                              hipStream_t stream) {
    const float* target_x = (const float*)d_in[0];   // [B, T]    f32
    const float* vals     = (const float*)d_in[1];   // [B, T, D] f32
    const int*   mask     = (const int*)  d_in[2];   // [B, T, D] i32
    float*       out      = (float*)d_out;           // [B, L, D+3] f32

    const size_t shmem = (size_t)L_ * sizeof(int);   // 84KB dynamic LDS / block
    hipLaunchKernelGGL(TRIPLETFORMER_pack_kernel,
                       dim3(B_), dim3(512), shmem, stream,
                       target_x, vals, mask, out);
}